// DeletionChannel_22445499089174
// MI455X (gfx1250) — compile-verified
//
#include <hip/hip_runtime.h>
#include <stdint.h>

// Problem constants (from reference): B=4096, L=128, V=64, P=0.1
#define Bsz 4096
#define Lsz 128
#define Vsz 64
#define KEEPF 0.9f   // (1 - P)

// ---- CDNA5 async global<->LDS DMA (gfx1250), guarded so host pass / older
// toolchains still compile. If absent, falls back to plain float4 copies.
#if defined(__AMDGCN__) &&                                              \
    __has_builtin(__builtin_amdgcn_global_load_async_to_lds_b128) &&    \
    __has_builtin(__builtin_amdgcn_global_store_async_from_lds_b128)
#define USE_ASYNC 1
#else
#define USE_ASYNC 0
#endif

#if USE_ASYNC
// Toolchain signature (from diagnostic): param1 is a generic
// 'int __attribute__((vector_size(16)))*' (flat pointer to the global addr).
typedef int v4i __attribute__((vector_size(4 * sizeof(int))));
// LDS side: build an addrspace(3) v4i* via the integer-cast pattern
// (InferAddressSpaces folds this back to the AS3 pointer at -O3). If the
// builtin's LDS param is generic, AS3 -> generic converts implicitly anyway.
typedef __attribute__((address_space(3))) v4i as3_v4i;

__device__ __forceinline__ v4i* gptr_cast(const void* p) {
  return (v4i*)(uintptr_t)p;
}
__device__ __forceinline__ as3_v4i* lds_cast(void* p) {
  return (as3_v4i*)(uintptr_t)p;
}
__device__ __forceinline__ void wait_async0() {
#if __has_builtin(__builtin_amdgcn_s_wait_asynccnt)
  __builtin_amdgcn_s_wait_asynccnt(0);
#else
  asm volatile("s_wait_asynccnt 0" ::: "memory");
#endif
}
#endif

__global__ __launch_bounds__(256)
void deletion_channel_kernel(const float* __restrict__ msg,
                             const float* __restrict__ probs,
                             const unsigned char* __restrict__ mask,  // jax bool -> 1B
                             float* __restrict__ out_m,
                             float* __restrict__ out_p) {
  const int b        = blockIdx.x;
  const int tid      = threadIdx.x;
  const size_t rowBase = (size_t)b * (Lsz * Vsz);

#if USE_ASYNC
  __shared__ float sMsg[Lsz * Vsz];   // 32 KB tile, filled by async DMA
#endif
  __shared__ int sDest[Lsz];
  __shared__ int sWaveSum[4];
  __shared__ int sNk;

#if USE_ASYNC
  // ---- Phase 1: async-stage this row's messages tile into LDS (ASYNCcnt).
  // 2048 x b128 transfers; each wave instruction moves 512 contiguous bytes.
  {
    const float* g = msg + rowBase;
#pragma unroll
    for (int i = 0; i < 8; ++i) {
      const int c = tid + 256 * i;  // float4 chunk id
      __builtin_amdgcn_global_load_async_to_lds_b128(
          gptr_cast(g + (size_t)c * 4), lds_cast(&sMsg[c * 4]), 0, 0);
    }
  }
#endif

  // ---- Phase 2: stream-compaction destinations via wave32 ballot prefix sum.
  {
    bool kept = false;
    if (tid < Lsz) kept = (mask[(size_t)b * Lsz + tid] == 0);
    const unsigned long long bal = __ballot(kept);
    const int lane = tid & 31;
    const int pre  = __popcll(bal & ((1ull << lane) - 1ull));
    if (tid < Lsz && lane == 0) sWaveSum[tid >> 5] = __popcll(bal);
    __syncthreads();
    if (tid < Lsz) {
      const int w = tid >> 5;
      int base = 0;
#pragma unroll
      for (int k = 0; k < 4; ++k) base += (k < w) ? sWaveSum[k] : 0;
      sDest[tid] = kept ? (base + pre) : -1;
    }
    if (tid == 0) sNk = sWaveSum[0] + sWaveSum[1] + sWaveSum[2] + sWaveSum[3];
  }

  // ---- Phase 3: probs renorm (overlaps the async loads).
  // 16 lanes per vocab row: float4 per lane, row sum via shfl_xor tree.
  {
    const int vec = tid & 15;
#pragma unroll
    for (int it = 0; it < 8; ++it) {
      const int l = (tid >> 4) + 16 * it;
      const float4 p = *(const float4*)(probs + rowBase + (size_t)l * Vsz + vec * 4);
      float s = p.x + p.y + p.z + p.w;
      s += __shfl_xor(s, 1);
      s += __shfl_xor(s, 2);
      s += __shfl_xor(s, 4);
      s += __shfl_xor(s, 8);          // full 64-entry row sum
      float4 q;
      q.x = KEEPF * p.x; q.y = KEEPF * p.y; q.z = KEEPF * p.z; q.w = KEEPF * p.w;
      if (vec == 0) q.x = 1.0f - KEEPF * (s - p.x);  // EOS absorbs remainder
      *(float4*)(out_p + rowBase + (size_t)l * Vsz + vec * 4) = q;
    }
  }

#if USE_ASYNC
  wait_async0();     // this wave's tile portion landed in LDS
#endif
  __syncthreads();   // full tile + sDest + sNk visible block-wide

  // ---- Phase 4: emit compacted rows (LDS -> global async DMA) + EOS tail.
  {
    const int vec = tid & 15;
    const int nk  = sNk;
#pragma unroll
    for (int it = 0; it < 8; ++it) {
      const int l = (tid >> 4) + 16 * it;
      const int d = sDest[l];
      if (d >= 0) {
#if USE_ASYNC
        __builtin_amdgcn_global_store_async_from_lds_b128(
            gptr_cast(out_m + rowBase + (size_t)d * Vsz + vec * 4),
            lds_cast(&sMsg[l * Vsz + vec * 4]), 0, 0);
#else
        *(float4*)(out_m + rowBase + (size_t)d * Vsz + vec * 4) =
            *(const float4*)(msg + rowBase + (size_t)l * Vsz + vec * 4);
#endif
      }
      if (l >= nk) {  // EOS one-hot tail rows
        float4 q;
        q.x = (vec == 0) ? 1.0f : 0.0f;
        q.y = 0.0f; q.z = 0.0f; q.w = 0.0f;
        *(float4*)(out_m + rowBase + (size_t)l * Vsz + vec * 4) = q;
      }
    }
  }

#if USE_ASYNC
  wait_async0();  // drain async stores (S_ENDPGM also implies wait-idle)
#endif
}

extern "C" void kernel_launch(void* const* d_in, const int* in_sizes, int n_in,
                              void* d_out, int out_size, void* d_ws, size_t ws_size,
                              hipStream_t stream) {
  (void)in_sizes; (void)n_in; (void)out_size; (void)d_ws; (void)ws_size;
  const float* msg            = (const float*)d_in[0];
  const float* probs          = (const float*)d_in[1];
  const unsigned char* mask   = (const unsigned char*)d_in[2];  // bool, 1B/elem
  float* out_m = (float*)d_out;                                  // [B,L,V]
  float* out_p = out_m + (size_t)Bsz * Lsz * Vsz;                // [B,L,V]
  hipLaunchKernelGGL(deletion_channel_kernel, dim3(Bsz), dim3(256), 0, stream,
                     msg, probs, mask, out_m, out_p);
}